// PointsSampler_23845658427861
// MI455X (gfx1250) — compile-verified
//
#include <hip/hip_runtime.h>
#include <stdint.h>
#include <stddef.h>

// Problem constants (match the reference).
#define B_      8
#define N_      65536
#define NPOINT_ 4096
#define THREADS 1024
#define NC      26624               // points staged in LDS = 13 * 2048
#define NCJ     13                  // LDS pair-chunks   (13 * 2048 = 26624 pts)
#define NGJ     19                  // global pair-chunks(19 * 2048 = 38912 pts)
#define CACHE_BYTES (NC * 12)       // 319488 B (SoA: x[NC], y[NC], z[NC])
#define LDS_BYTES   (CACHE_BYTES + 128 + 128 + 16)

// ---- CDNA5 TDM availability ------------------------------------------------
#if defined(__HIP_DEVICE_COMPILE__) && defined(__gfx1250__)
#if __has_builtin(__builtin_amdgcn_tensor_load_to_lds) && \
    __has_builtin(__builtin_amdgcn_s_wait_tensorcnt)
#define USE_TDM 1
#endif
#endif
#ifndef USE_TDM
#define USE_TDM 0
#endif

#if __has_include(<hip/amd_detail/amd_gfx1250_TDM.h>)
#define TDM_SIX_ARG 1   // therock-10.0 headers -> 6-arg builtin
#else
#define TDM_SIX_ARG 0   // ROCm 7.2 -> 5-arg builtin (confirmed active last round)
#endif

typedef unsigned int u32x4 __attribute__((ext_vector_type(4)));
typedef int          i32x4 __attribute__((ext_vector_type(4)));
typedef int          i32x8 __attribute__((ext_vector_type(8)));
typedef float        f32x2 __attribute__((ext_vector_type(2)));

#if USE_TDM
// Strided-gather TDM: copy NC dwords from gsrc with a stride of 3 dwords into
// LDS at byte offset lds_off (AoS -> SoA plane extraction done by the DMA).
// D#: tile 1 x NC, tensor_dim0=1, tensor_dim0_stride=3, data_size=4B.
__device__ __forceinline__ void tdm_gather_stride3(uint32_t lds_off,
                                                   const float* gsrc) {
    const uint64_t ga = (uint64_t)(uintptr_t)gsrc;
    u32x4 g0;
    g0[0] = 1u;                                      // count=1, user descriptor
    g0[1] = lds_off;                                 // lds_addr (bytes)
    g0[2] = (uint32_t)ga;                            // global_addr[31:0]
    g0[3] = (uint32_t)((ga >> 32) & 0x01FFFFFFu)     // global_addr[56:32]
          | (2u << 30);                              // type=2 ("image")
    i32x8 g1;
    g1[0] = (int)(2u << 16);                         // data_size=4B, mask=0
    g1[1] = (int)(1u << 16);                         // tensor_dim0 = 1 (lo16)
    g1[2] = (int)((NC & 0xFFFFu) << 16);             // tensor_dim1 = NC (lo16)
    g1[3] = (int)(1u << 16);                         // tile_dim0 = 1
    g1[4] = (int)(NC & 0xFFFFu);                     // tile_dim1 = NC, tile2=0
    g1[5] = 3;                                       // tensor_dim0_stride = 3
    g1[6] = 0;
    g1[7] = 0;
    i32x4 z4 = {0, 0, 0, 0};
#if TDM_SIX_ARG
    i32x8 z8 = {0, 0, 0, 0, 0, 0, 0, 0};
    __builtin_amdgcn_tensor_load_to_lds(g0, g1, z4, z4, z8, 0);
#else
    __builtin_amdgcn_tensor_load_to_lds(g0, g1, z4, z4, 0);
#endif
}
#endif

// Packed distance update for a pair of consecutive point indices (iB, iB+1).
// v_pk_add/v_pk_mul/v_pk_fma_f32 candidates; element 0 tested first so the
// lowest index wins exact-value ties (matches jnp.argmax first-occurrence).
__device__ __forceinline__ void pair_step(f32x2 xx, f32x2 yy, f32x2 zz,
                                          f32x2 qx2, f32x2 qy2, f32x2 qz2,
                                          float& s0, float& s1, int iB,
                                          float& bv, int& bi) {
    const f32x2 dx = xx - qx2, dy = yy - qy2, dz = zz - qz2;
    const f32x2 acc = __builtin_elementwise_fma(
        dx, dx, __builtin_elementwise_fma(dy, dy, dz * dz));
    const float n0 = fminf(s0, acc.x); s0 = n0;
    const float n1 = fminf(s1, acc.y); s1 = n1;
    if (n0 > bv) { bv = n0; bi = iB; }
    if (n1 > bv) { bv = n1; bi = iB + 1; }
}

__device__ __forceinline__ void amax2(float& v, int& i, float ov, int oi) {
    if (ov > v || (ov == v && oi < i)) { v = ov; i = oi; }
}

__global__ __launch_bounds__(THREADS)
void fps_kernel(const float* __restrict__ pts, int* __restrict__ out)
{
    extern __shared__ char smem[];
    float* xs   = (float*)smem;              // [NC] x-plane
    float* ys   = xs + NC;                   // [NC] y-plane
    float* zs   = ys + NC;                   // [NC] z-plane
    float* redv = (float*)(smem + CACHE_BYTES);
    int*   redi = (int*)  (smem + CACHE_BYTES + 128);
    float* bcast= (float*)(smem + CACHE_BYTES + 256);

    const int b = blockIdx.x;
    const int t = threadIdx.x;
    const float* __restrict__ src = pts + (size_t)b * N_ * 3;

    // ---- Stage first NC points into LDS as SoA ----------------------------
#if USE_TDM
    if (t < 32) {  // wave 0 only; TDM ops are in-order with each other
        tdm_gather_stride3((uint32_t)(uintptr_t)xs, src + 0);
        tdm_gather_stride3((uint32_t)(uintptr_t)ys, src + 1);
        tdm_gather_stride3((uint32_t)(uintptr_t)zs, src + 2);
        __builtin_amdgcn_s_wait_tensorcnt(0);
    }
#else
    for (int i = t; i < NC; i += THREADS) {
        xs[i] = src[3 * i + 0];
        ys[i] = src[3 * i + 1];
        zs[i] = src[3 * i + 2];
    }
#endif
    if (t == 0) out[(size_t)b * NPOINT_] = 0;   // first sample is index 0
    __syncthreads();

    // Persistent min-distance state: 64 floats per lane, in VGPRs.
    float dist[2 * (NCJ + NGJ)];
#pragma unroll
    for (int j = 0; j < 2 * (NCJ + NGJ); ++j) dist[j] = 1e10f;

    float qx = src[0], qy = src[1], qz = src[2];

    for (int s = 1; s < NPOINT_; ++s) {
        float bv = -1.0f;
        int   bi = 0x7FFFFFFF;
        const f32x2 qx2 = {qx, qx}, qy2 = {qy, qy}, qz2 = {qz, qz};

        // LDS-resident pairs: conflict-free ds_load_b64 per plane.
#pragma unroll
        for (int jc = 0; jc < NCJ; ++jc) {
            const int iB = jc * 2048 + 2 * t;
            const f32x2 xx = *(const f32x2*)(xs + iB);
            const f32x2 yy = *(const f32x2*)(ys + iB);
            const f32x2 zz = *(const f32x2*)(zs + iB);
            pair_step(xx, yy, zz, qx2, qy2, qz2,
                      dist[2 * jc], dist[2 * jc + 1], iB, bv, bi);
        }
        // L2-resident pairs: 24 contiguous bytes/lane as 3 aligned b64 loads.
#pragma unroll
        for (int jc = 0; jc < NGJ; ++jc) {
            const int iB = NC + jc * 2048 + 2 * t;
            const float* gp = src + (size_t)iB * 3;
            const f32x2 a = *(const f32x2*)(gp + 0);   // x0 y0
            const f32x2 c = *(const f32x2*)(gp + 2);   // z0 x1
            const f32x2 e = *(const f32x2*)(gp + 4);   // y1 z1
            const f32x2 xx = {a.x, c.y};
            const f32x2 yy = {a.y, e.x};
            const f32x2 zz = {c.x, e.y};
            pair_step(xx, yy, zz, qx2, qy2, qz2,
                      dist[2 * NCJ + 2 * jc], dist[2 * NCJ + 2 * jc + 1],
                      iB, bv, bi);
        }

        // wave32 butterfly argmax reduction
#pragma unroll
        for (int off = 16; off; off >>= 1) {
            const float ov = __shfl_xor(bv, off, 32);
            const int   oi = __shfl_xor(bi, off, 32);
            amax2(bv, bi, ov, oi);
        }
        const int wave = t >> 5, lane = t & 31;
        if (lane == 0) { redv[wave] = bv; redi[wave] = bi; }
        __syncthreads();

        if (wave == 0) {
            bv = redv[lane]; bi = redi[lane];
#pragma unroll
            for (int off = 16; off; off >>= 1) {
                const float ov = __shfl_xor(bv, off, 32);
                const int   oi = __shfl_xor(bi, off, 32);
                amax2(bv, bi, ov, oi);
            }
            if (lane == 0) {
                out[(size_t)b * NPOINT_ + s] = bi;
                if (bi < NC) {
                    bcast[0] = xs[bi]; bcast[1] = ys[bi]; bcast[2] = zs[bi];
                } else {
                    const float* p = src + (size_t)bi * 3;
                    bcast[0] = p[0]; bcast[1] = p[1]; bcast[2] = p[2];
                }
            }
        }
        __syncthreads();
        qx = bcast[0]; qy = bcast[1]; qz = bcast[2];
        // bcast is only rewritten after the next iteration's first
        // __syncthreads(), so these reads are race-free.
    }
}

extern "C" void kernel_launch(void* const* d_in, const int* in_sizes, int n_in,
                              void* d_out, int out_size, void* d_ws, size_t ws_size,
                              hipStream_t stream) {
    (void)in_sizes; (void)n_in; (void)out_size; (void)d_ws; (void)ws_size;
    const float* pts = (const float*)d_in[0];   // (B, N, 3) float32
    // d_in[1] (features) is ignored by D-FPS, matching the reference.
    int* out = (int*)d_out;                     // (B, NPOINT) int32 indices

    (void)hipFuncSetAttribute((const void*)fps_kernel,
                              hipFuncAttributeMaxDynamicSharedMemorySize,
                              LDS_BYTES);
    fps_kernel<<<dim3(B_), dim3(THREADS), LDS_BYTES, stream>>>(pts, out);
}